// ResonantFFN_16827681866028
// MI455X (gfx1250) — compile-verified
//
#include <hip/hip_runtime.h>
#include <hip/hip_bf16.h>

// ---- problem constants (match reference) ----
#define BB 2
#define LL 512
#define DD 512
#define PP 64
#define HH 2048
#define BL (BB * LL)   // 1024

typedef __attribute__((ext_vector_type(2))) float v2f;
typedef __attribute__((ext_vector_type(8))) float v8f;

// ---------------------------------------------------------------------------
// Gate kernel: gate[row,h] = (1/sqrt(P)) * sum_p cos( (x_imag[row,p]*pos_freq[l,p]) * w[p,h] + b[p,h] )
// Block: 256 threads; tile = 8 rows x 128 h. tc staged in LDS (broadcast reads).
// Each thread computes 4 rows for one h -> each w/b load amortized over 4 cos.
// ---------------------------------------------------------------------------
__global__ void gate_kernel(const float* __restrict__ x_imag,
                            const float* __restrict__ pos_freq,
                            const float* __restrict__ w,
                            const float* __restrict__ b,
                            float* __restrict__ gate) {
    __shared__ float tc[8][PP];
    const int hb  = blockIdx.x;          // H / 128 tiles
    const int rb  = blockIdx.y;          // BL / 8 tiles
    const int tid = threadIdx.x;         // 256

    // stage time_content for 8 rows (8*64 = 512 elements)
    for (int i = tid; i < 8 * PP; i += 256) {
        const int r = i >> 6;
        const int p = i & (PP - 1);
        const int row = rb * 8 + r;
        const int l = row & (LL - 1);    // row % L
        tc[r][p] = x_imag[(size_t)row * PP + p] * pos_freq[(size_t)l * PP + p];
    }
    __syncthreads();

    const int h  = hb * 128 + (tid & 127);
    const int r0 = (tid >> 7) * 4;       // 0 or 4

    float a0 = 0.f, a1 = 0.f, a2 = 0.f, a3 = 0.f;
    #pragma unroll 4
    for (int p = 0; p < PP; ++p) {
        const float wv = w[(size_t)p * HH + h];
        const float bv = b[(size_t)p * HH + h];
        a0 += __cosf(fmaf(tc[r0 + 0][p], wv, bv));
        a1 += __cosf(fmaf(tc[r0 + 1][p], wv, bv));
        a2 += __cosf(fmaf(tc[r0 + 2][p], wv, bv));
        a3 += __cosf(fmaf(tc[r0 + 3][p], wv, bv));
    }
    const float scale = 0.125f;          // 1/sqrt(64)
    const int row = rb * 8 + r0;
    gate[(size_t)(row + 0) * HH + h] = a0 * scale;
    gate[(size_t)(row + 1) * HH + h] = a1 * scale;
    gate[(size_t)(row + 2) * HH + h] = a2 * scale;
    gate[(size_t)(row + 3) * HH + h] = a3 * scale;
}

// ---------------------------------------------------------------------------
// Generic fp32 WMMA GEMM: C = A(MxK, row-major) * B(KxN, row-major)
// optionally C *= gateMul (elementwise, MxN), optionally C += residual (MxN).
// One 16x16 C tile per wave via V_WMMA_F32_16X16X4_F32 (K advances 4/instr).
// A layout (16x4 f32): lanes 0-15 -> K=0,1 ; lanes 16-31 -> K=2,3 ; row = lane&15.
// C/D layout (8 VGPRs): vgpr r: lanes 0-15 -> M=r, lanes 16-31 -> M=8+r; N = lane&15.
// ---------------------------------------------------------------------------
__global__ void wmma_gemm_f32(const float* __restrict__ A,
                              const float* __restrict__ Bm,
                              const float* __restrict__ residual, // may be null
                              const float* __restrict__ gateMul,  // may be null
                              float* __restrict__ C,
                              int M, int N, int K) {
    const int lane   = threadIdx.x & 31;
    const int wid    = threadIdx.x >> 5;
    const int tilesN = N >> 4;
    const int tiles  = (M >> 4) * tilesN;
    const int tile   = blockIdx.x * (blockDim.x >> 5) + wid;
    if (tile >= tiles) return;           // wave-uniform: EXEC stays all-ones

    const int mt = tile / tilesN;
    const int nt = tile - mt * tilesN;

    const int halfsel = lane >> 4;       // 0 or 1
    const int kb   = halfsel * 2;        // K sub-offset held by this lane
    const int arow = mt * 16 + (lane & 15);
    const int bcol = nt * 16 + (lane & 15);

    const float* Ap = A  + (size_t)arow * K + kb;
    const float* Bp = Bm + (size_t)kb * N + bcol;

    v8f acc = {0.f, 0.f, 0.f, 0.f, 0.f, 0.f, 0.f, 0.f};

    #pragma unroll 4
    for (int k = 0; k < K; k += 4) {
        v2f av;
        av.x = Ap[k];
        av.y = Ap[k + 1];
        v2f bv;
        bv.x = Bp[(size_t)k * N];
        bv.y = Bp[(size_t)(k + 1) * N];
        acc = __builtin_amdgcn_wmma_f32_16x16x4_f32(
            /*neg_a=*/false, av, /*neg_b=*/false, bv,
            /*c_mod=*/(short)0, acc, /*reuse_a=*/false, /*reuse_b=*/false);
    }

    const int srow = mt * 16 + halfsel * 8;
    const int scol = bcol;
    #pragma unroll
    for (int r = 0; r < 8; ++r) {
        const size_t idx = (size_t)(srow + r) * N + scol;
        float v = acc[r];
        if (gateMul)  v *= gateMul[idx];
        if (residual) v += residual[idx];
        C[idx] = v;
    }
}

extern "C" void kernel_launch(void* const* d_in, const int* in_sizes, int n_in,
                              void* d_out, int out_size, void* d_ws, size_t ws_size,
                              hipStream_t stream) {
    (void)in_sizes; (void)n_in; (void)out_size; (void)ws_size;

    const float* x_real      = (const float*)d_in[0]; // (B,L,D)
    const float* x_imag      = (const float*)d_in[1]; // (B,L,P)
    const float* pos_freq    = (const float*)d_in[2]; // (L,P)
    const float* W_up        = (const float*)d_in[3]; // (D,H)
    const float* w           = (const float*)d_in[4]; // (P,H)
    const float* b           = (const float*)d_in[5]; // (P,H)
    const float* W_down_real = (const float*)d_in[6]; // (H,D)
    const float* W_down_imag = (const float*)d_in[7]; // (H,P)

    float* out_real = (float*)d_out;                       // (B,L,D)
    float* out_imag = out_real + (size_t)BL * DD;          // (B,L,P)

    float* gate   = (float*)d_ws;                          // (BL,H) = 8 MB
    float* outbuf = gate + (size_t)BL * HH;                // (BL,H) = 8 MB

    // 1) gate (transcendental-bound: 134M v_cos_f32)
    gate_kernel<<<dim3(HH / 128, BL / 8), 256, 0, stream>>>(
        x_imag, pos_freq, w, b, gate);

    // 2) out = (x_real @ W_up) * gate      [M=1024, N=2048, K=512] -> 8192 tiles
    wmma_gemm_f32<<<(BL / 16) * (HH / 16) / 8, 256, 0, stream>>>(
        x_real, W_up, /*residual=*/nullptr, /*gateMul=*/gate, outbuf, BL, HH, DD);

    // 3) out_real = x_real + out @ W_down_real   [M=1024, N=512, K=2048] -> 2048 tiles
    wmma_gemm_f32<<<(BL / 16) * (DD / 16) / 8, 256, 0, stream>>>(
        outbuf, W_down_real, /*residual=*/x_real, /*gateMul=*/nullptr, out_real, BL, DD, HH);

    // 4) out_imag = x_imag + gate @ W_down_imag  [M=1024, N=64, K=2048] -> 256 tiles
    wmma_gemm_f32<<<(BL / 16) * (PP / 16) / 8, 256, 0, stream>>>(
        gate, W_down_imag, /*residual=*/x_imag, /*gateMul=*/nullptr, out_imag, BL, PP, HH);
}